// BoxSampler_6674379178401
// MI455X (gfx1250) — compile-verified
//
#include <hip/hip_runtime.h>
#include <hip/hip_bf16.h>

#define N_ANCHORS   250000
#define NUM_CLASSES 80
#define IMG_W       512.0f
#define IMG_H       512.0f
#define K_PRE       2000
#define K_PAD       2048
#define MAX_DET     300
#define NUM_WORDS   63            // ceil(2000/32)
#define SCORE_THRESH 0.05f

typedef __attribute__((ext_vector_type(2))) float v2f;
typedef __attribute__((ext_vector_type(8))) float v8f;

// ---------------- workspace layout (bytes) ----------------
static constexpr size_t OFF_BOXES   = 0;                         // 250000 * float4
static constexpr size_t OFF_SCORES  = 4000000;                   // 250000 * f32
static constexpr size_t OFF_HIST    = 5000000;                   // 256 * u32
static constexpr size_t OFF_STATE   = 5001024;                   // 8 * u32
static constexpr size_t OFF_CNTRS   = 5001088;                   // 2 * u32 (padded)
static constexpr size_t OFF_CKEY    = 5001152;                   // 2048 * u32
static constexpr size_t OFF_CIDX    = 5009344;                   // 2048 * u32
static constexpr size_t OFF_TOPI    = 5017536;                   // 2000 * u32
static constexpr size_t OFF_BOXK    = 5025536;                   // 2000 * float4 (16B aligned)
static constexpr size_t OFF_MASK    = 5057536;                   // 2000 * 63 * u32
static constexpr size_t OFF_DET     = 5561536;                   // 1 + 300 ints

// ---------------- init ----------------
__global__ void bs_init_kernel(unsigned* hist, unsigned* state, unsigned* counters,
                               unsigned* cand_key, unsigned* cand_idx, int* det) {
  int t = blockIdx.x * blockDim.x + threadIdx.x;
  if (t < 256) hist[t] = 0u;
  if (t < 8)   state[t] = (t == 1) ? (unsigned)K_PRE : 0u;
  if (t < 2)   counters[t] = 0u;
  if (t < K_PAD) { cand_key[t] = 0u; cand_idx[t] = 0u; }
  if (t < MAX_DET + 1) det[t] = 0;
}

// ---------------- decode boxes + per-anchor max class score ----------------
__global__ __launch_bounds__(256) void bs_decode_score_kernel(
    const float4* __restrict__ anchors, const float4* __restrict__ regression,
    const float4* __restrict__ cls4, float4* __restrict__ boxes_ws,
    float* __restrict__ scores_ws) {
  int a = blockIdx.x * blockDim.x + threadIdx.x;
  if (a >= N_ANCHORS) return;
  float4 an = anchors[a];
  float4 rg = regression[a];
  float w = an.z - an.x, h = an.w - an.y;
  float cx = an.x + 0.5f * w, cy = an.y + 0.5f * h;
  float pcx = cx + rg.x * 0.1f * w;
  float pcy = cy + rg.y * 0.1f * h;
  float pw = __expf(rg.z * 0.2f) * w;
  float ph = __expf(rg.w * 0.2f) * h;
  float4 box;
  box.x = fmaxf(pcx - 0.5f * pw, 0.0f);
  box.y = fmaxf(pcy - 0.5f * ph, 0.0f);
  box.z = fminf(pcx + 0.5f * pw, IMG_W);
  box.w = fminf(pcy + 0.5f * ph, IMG_H);
  boxes_ws[a] = box;
  const float4* row = cls4 + (size_t)a * (NUM_CLASSES / 4);
  float m = -1.0f;
#pragma unroll
  for (int k = 0; k < NUM_CLASSES / 4; ++k) {
    float4 v = row[k];
    m = fmaxf(m, fmaxf(fmaxf(v.x, v.y), fmaxf(v.z, v.w)));
  }
  scores_ws[a] = (m > SCORE_THRESH) ? m : -__builtin_inff();
}

// ---------------- radix-select pass: byte histogram ----------------
__global__ __launch_bounds__(256) void bs_hist_kernel(const float* __restrict__ scores,
                                                      unsigned* __restrict__ hist,
                                                      const unsigned* __restrict__ state, int b) {
  int i = blockIdx.x * blockDim.x + threadIdx.x;
  if (i >= N_ANCHORS) return;
  unsigned key = __float_as_uint(scores[i]);
  if (key & 0x80000000u) return;  // filtered (-inf) or non-positive
  bool part = (b == 3) || ((key >> ((b + 1) * 8)) == state[0]);
  if (part) atomicAdd(&hist[(key >> (b * 8)) & 255u], 1u);
}

// ---------------- radix-select pass: pick threshold byte ----------------
__global__ __launch_bounds__(256) void bs_select_kernel(unsigned* hist, unsigned* state, int b) {
  __shared__ unsigned cnt_s[256];
  int t = threadIdx.x;
  cnt_s[t] = hist[t];
  __syncthreads();
  if (t == 0) {
    unsigned remaining = state[1];
    unsigned cum = 0;
    int c = 0;
    for (int i = 255; i >= 0; --i) {
      unsigned ci = cnt_s[i];
      if (cum + ci >= remaining) { c = i; break; }
      cum += ci;
    }
    state[0] = (state[0] << 8) | (unsigned)c;  // extend prefix
    state[2] += cum;                            // strictly-above count so far
    state[1] = remaining - cum;                 // take this many among == prefix
    if (b == 0) {
      state[3] = state[0];   // final 32-bit threshold key
      state[4] = state[2];   // count strictly above threshold
      state[5] = state[1];   // number to take among equal
    }
  }
  __syncthreads();
  hist[t] = 0u;              // ready for next pass
}

// ---------------- compact candidates above/equal threshold ----------------
__global__ __launch_bounds__(256) void bs_compact_kernel(const float* __restrict__ scores,
                                                         const unsigned* __restrict__ state,
                                                         unsigned* counters,
                                                         unsigned* cand_key, unsigned* cand_idx) {
  int i = blockIdx.x * blockDim.x + threadIdx.x;
  if (i >= N_ANCHORS) return;
  unsigned key = __float_as_uint(scores[i]);
  if (key & 0x80000000u) return;
  unsigned th = state[3];
  if (key > th) {
    unsigned p = atomicAdd(&counters[0], 1u);
    if (p < K_PRE) { cand_key[p] = key; cand_idx[p] = (unsigned)i; }
  } else if (key == th) {
    unsigned q = atomicAdd(&counters[1], 1u);
    if (q < state[5]) {
      unsigned p = state[4] + q;
      if (p < K_PRE) { cand_key[p] = key; cand_idx[p] = (unsigned)i; }
    }
  }
}

// ---------------- bitonic sort (desc by score bits, asc by idx) + gather ----------------
__global__ __launch_bounds__(1024) void bs_sort_gather_kernel(
    unsigned* cand_key, unsigned* cand_idx, unsigned* __restrict__ top_i,
    float4* __restrict__ boxes_k, const float4* __restrict__ boxes_ws) {
  __shared__ unsigned long long d[K_PAD];
  int t = threadIdx.x;
  for (int i = t; i < K_PAD; i += 1024)
    d[i] = ((unsigned long long)cand_key[i] << 32) | (unsigned)(~cand_idx[i]);
  __syncthreads();
  for (int k = 2; k <= K_PAD; k <<= 1) {
    for (int j = k >> 1; j > 0; j >>= 1) {
      for (int i = t; i < K_PAD; i += 1024) {
        int ixj = i ^ j;
        if (ixj > i) {
          bool desc = ((i & k) == 0);
          unsigned long long a = d[i], b = d[ixj];
          if ((a < b) == desc) { d[i] = b; d[ixj] = a; }
        }
      }
      __syncthreads();
    }
  }
  for (int i = t; i < K_PRE; i += 1024) {
    unsigned key = (unsigned)(d[i] >> 32);
    unsigned idx = ~(unsigned)(d[i] & 0xffffffffu);
    cand_key[i] = key;        // sorted keys back (validity for NMS)
    cand_idx[i] = idx;
    top_i[i] = idx;
    boxes_k[i] = boxes_ws[idx];
  }
}

// ---------------- NMS suppression-mask: WMMA area-sums + VALU intersection ----------------
// Each wave handles a 16-row x 32-col block. Two 16x16 WMMA tiles compute
// areaSum[m][n] = area_row[m] + area_col[n] on the matrix pipe; VALU computes
// intersections; ballot_w32 packs two rows of suppress bits per C register.
__global__ __launch_bounds__(32) void bs_nms_mask_kernel(const float* __restrict__ boxes_k,
                                                         unsigned* __restrict__ mask) {
  __shared__ float rb[16][4];
  const int lane = threadIdx.x;
  const int cw = blockIdx.x;        // col word 0..62
  const int rt = blockIdx.y;        // row tile 0..124
  const int r0 = rt * 16;
  const int c0 = cw * 32;
  if (lane < 16) {
    const float4 b = *(const float4*)(boxes_k + (size_t)(r0 + lane) * 4);
    rb[lane][0] = b.x; rb[lane][1] = b.y; rb[lane][2] = b.z; rb[lane][3] = b.w;
  }
  __syncthreads();
  const int halfsel = lane >> 4;    // 0: lanes 0-15, 1: lanes 16-31
  const int l15 = lane & 15;
  // A (16x4 f32): VGPR0 = K0 (area_row) / K2 (0); VGPR1 = K1 (1.0) / K3 (0)
  float area_r = 0.0f, one_k = 0.0f;
  if (halfsel == 0) {
    area_r = (rb[l15][2] - rb[l15][0]) * (rb[l15][3] - rb[l15][1]);
    one_k = 1.0f;
  }
  v2f A; A.x = area_r; A.y = one_k;

  unsigned words[16];
#pragma unroll
  for (int v = 0; v < 16; ++v) words[v] = 0u;

  for (int h = 0; h < 2; ++h) {
    int col = c0 + h * 16 + l15;
    int colc = (col < K_PRE) ? col : (K_PRE - 1);
    float4 cb = *(const float4*)(boxes_k + (size_t)colc * 4);
    float area_c = (cb.z - cb.x) * (cb.w - cb.y);
    // B (4x16 f32): VGPR0 = K0 row (1.0) / K2 (0); VGPR1 = K1 row (area_col) / K3 (0)
    v2f B;
    B.x = (halfsel == 0) ? 1.0f : 0.0f;
    B.y = (halfsel == 0) ? area_c : 0.0f;
    v8f C = {0.f, 0.f, 0.f, 0.f, 0.f, 0.f, 0.f, 0.f};
    C = __builtin_amdgcn_wmma_f32_16x16x4_f32(false, A, false, B, (short)0, C, false, false);
#pragma unroll
    for (int v = 0; v < 8; ++v) {
      int m = v + halfsel * 8;
      float ix1 = fmaxf(rb[m][0], cb.x);
      float iy1 = fmaxf(rb[m][1], cb.y);
      float ix2 = fminf(rb[m][2], cb.z);
      float iy2 = fminf(rb[m][3], cb.w);
      float inter = fmaxf(ix2 - ix1, 0.0f) * fmaxf(iy2 - iy1, 0.0f);
      int gr = r0 + m;
      // iou > 0.1  <=>  inter > 0.1*(areaSum - inter)  <=>  11*inter > areaSum
      bool sup = (11.0f * inter > C[v]) && (col > gr) && (col < K_PRE);
      unsigned bal = __builtin_amdgcn_ballot_w32(sup);
      words[v]     |= (bal & 0xFFFFu) << (16 * h);   // rows m = v   (lanes 0-15)
      words[v + 8] |= (bal >> 16)     << (16 * h);   // rows m = v+8 (lanes 16-31)
    }
  }
  unsigned w = 0u;
#pragma unroll
  for (int v = 0; v < 16; ++v)
    if (lane == v) w = words[v];
  if (lane < 16) mask[(size_t)(r0 + lane) * NUM_WORDS + cw] = w;
}

// ---------------- serial greedy NMS reduce: single wave32, no barriers ----------------
__global__ __launch_bounds__(32) void bs_nms_reduce_kernel(const unsigned* __restrict__ mask,
                                                           const unsigned* __restrict__ cand_key,
                                                           int* __restrict__ det) {
  __shared__ unsigned kw_lds[NUM_WORDS];
  int lane = threadIdx.x;
  int w0 = lane, w1 = lane + 32;
  unsigned k0 = 0u, k1 = 0u;
  for (int j = 0; j < 32; ++j) {
    int i0 = w0 * 32 + j;
    if (i0 < K_PRE && cand_key[i0] != 0u) k0 |= (1u << j);
    if (w1 < NUM_WORDS) {
      int i1 = w1 * 32 + j;
      if (i1 < K_PRE && cand_key[i1] != 0u) k1 |= (1u << j);
    }
  }
  for (int i = 0; i < K_PRE; ++i) {
    __builtin_prefetch(mask + (size_t)((i + 8 < K_PRE) ? i + 8 : i) * NUM_WORDS + w0, 0, 0);
    int w = i >> 5;
    unsigned kw = (w < 32) ? __shfl(k0, w, 32) : __shfl(k1, w - 32, 32);
    if (kw & (1u << (i & 31))) {
      unsigned m0 = mask[(size_t)i * NUM_WORDS + w0];
      unsigned m1 = (w1 < NUM_WORDS) ? mask[(size_t)i * NUM_WORDS + w1] : 0u;
      k0 &= ~m0;
      k1 &= ~m1;
    }
  }
  kw_lds[w0] = k0;
  if (w1 < NUM_WORDS) kw_lds[w1] = k1;
  __syncthreads();
  if (lane == 0) {
    int cnt = 0;
    for (int i = 0; i < K_PRE && cnt < MAX_DET; ++i)
      if (kw_lds[i >> 5] & (1u << (i & 31))) det[1 + cnt++] = i;
    det[0] = cnt;
  }
}

// ---------------- emit final detections ----------------
__global__ __launch_bounds__(128) void bs_output_kernel(const int* __restrict__ det,
                                                        const unsigned* __restrict__ top_i,
                                                        const float* __restrict__ boxes_k,
                                                        const float* __restrict__ cls,
                                                        float* __restrict__ out) {
  int d = blockIdx.x * blockDim.x + threadIdx.x;
  if (d >= MAX_DET) return;
  int cnt = det[0];
  if (d < cnt) {
    int slot = det[1 + d];
    unsigned ai = top_i[slot];
    const float* row = cls + (size_t)ai * NUM_CLASSES;
    float best = row[0];
    int bc = 0;
    for (int k = 1; k < NUM_CLASSES; ++k) {
      float v = row[k];
      if (v > best) { best = v; bc = k; }
    }
    out[d] = best;                                   // nms_scores
    out[MAX_DET + d] = (float)bc;                    // classes
    out[2 * MAX_DET + d * 4 + 0] = boxes_k[slot * 4 + 0];
    out[2 * MAX_DET + d * 4 + 1] = boxes_k[slot * 4 + 1];
    out[2 * MAX_DET + d * 4 + 2] = boxes_k[slot * 4 + 2];
    out[2 * MAX_DET + d * 4 + 3] = boxes_k[slot * 4 + 3];
    out[6 * MAX_DET + d] = (float)ai;                // selected_indices
    out[7 * MAX_DET + d] = 1.0f;                     // valid
  } else {
    out[d] = 0.0f;
    out[MAX_DET + d] = 0.0f;
    out[2 * MAX_DET + d * 4 + 0] = 0.0f;
    out[2 * MAX_DET + d * 4 + 1] = 0.0f;
    out[2 * MAX_DET + d * 4 + 2] = 0.0f;
    out[2 * MAX_DET + d * 4 + 3] = 0.0f;
    out[6 * MAX_DET + d] = -1.0f;
    out[7 * MAX_DET + d] = 0.0f;
  }
}

extern "C" void kernel_launch(void* const* d_in, const int* in_sizes, int n_in,
                              void* d_out, int out_size, void* d_ws, size_t ws_size,
                              hipStream_t stream) {
  const float4* anchors    = (const float4*)d_in[1];
  const float4* regression = (const float4*)d_in[2];
  const float*  cls        = (const float*)d_in[3];
  const float4* cls4       = (const float4*)d_in[3];
  float* out = (float*)d_out;

  char* ws = (char*)d_ws;
  float4*   boxes_ws = (float4*)(ws + OFF_BOXES);
  float*    scores   = (float*)(ws + OFF_SCORES);
  unsigned* hist     = (unsigned*)(ws + OFF_HIST);
  unsigned* state    = (unsigned*)(ws + OFF_STATE);
  unsigned* counters = (unsigned*)(ws + OFF_CNTRS);
  unsigned* cand_key = (unsigned*)(ws + OFF_CKEY);
  unsigned* cand_idx = (unsigned*)(ws + OFF_CIDX);
  unsigned* top_i    = (unsigned*)(ws + OFF_TOPI);
  float4*   boxes_k4 = (float4*)(ws + OFF_BOXK);
  float*    boxes_k  = (float*)(ws + OFF_BOXK);
  unsigned* mask     = (unsigned*)(ws + OFF_MASK);
  int*      det      = (int*)(ws + OFF_DET);

  const int nblk = (N_ANCHORS + 255) / 256;

  bs_init_kernel<<<(K_PAD + 255) / 256, 256, 0, stream>>>(hist, state, counters,
                                                          cand_key, cand_idx, det);
  bs_decode_score_kernel<<<nblk, 256, 0, stream>>>(anchors, regression, cls4,
                                                   boxes_ws, scores);
  for (int b = 3; b >= 0; --b) {
    bs_hist_kernel<<<nblk, 256, 0, stream>>>(scores, hist, state, b);
    bs_select_kernel<<<1, 256, 0, stream>>>(hist, state, b);
  }
  bs_compact_kernel<<<nblk, 256, 0, stream>>>(scores, state, counters, cand_key, cand_idx);
  bs_sort_gather_kernel<<<1, 1024, 0, stream>>>(cand_key, cand_idx, top_i, boxes_k4, boxes_ws);

  dim3 mgrid(NUM_WORDS, K_PRE / 16);   // 63 x 125
  bs_nms_mask_kernel<<<mgrid, 32, 0, stream>>>(boxes_k, mask);
  bs_nms_reduce_kernel<<<1, 32, 0, stream>>>(mask, cand_key, det);
  bs_output_kernel<<<(MAX_DET + 127) / 128, 128, 0, stream>>>(det, top_i, boxes_k, cls, out);
}